// OptNet_47244640256490
// MI455X (gfx1250) — compile-verified
//
#include <hip/hip_runtime.h>
#include <math.h>

typedef __attribute__((ext_vector_type(2))) float v2f;
typedef __attribute__((ext_vector_type(8))) float v8f;

#define BATCH 8192
#define FEAT  784
#define HID   2048
#define NCLS  10
#define NINEQ 200
#define BN_EPS 1e-5f
#define QP_EPS 1e-4f
#define QP_ITERS 30
#define NPART 4
#define MPART (NINEQ / NPART)   /* 50 constraints per wave */

// ---------------------------------------------------------------------------
// FC1: y1 = relu(x @ fc1_w^T + fc1_b)   [8192,784]x[2048,784]^T -> [8192,2048]
// fp32 WMMA 16x16x4. One wave computes a 16x64 output tile (4 N-subtiles).
// ---------------------------------------------------------------------------
__global__ __launch_bounds__(32) void fc1_wmma(const float* __restrict__ x,
                                               const float* __restrict__ w,
                                               const float* __restrict__ bias,
                                               float* __restrict__ y1) {
  const int lane = threadIdx.x;
  const int half = lane >> 4;
  const int l    = lane & 15;
  const int row0 = blockIdx.x * 16;
  const int col0 = blockIdx.y * 64;

  v8f acc[4] = {};
  const float* xrow = x + (size_t)(row0 + l) * FEAT + 2 * half;
  const float* wcol = w + (size_t)(col0 + l) * FEAT + 2 * half;

  for (int k = 0; k < FEAT; k += 4) {
    v2f a  = *(const v2f*)(xrow + k);
    v2f b0 = *(const v2f*)(wcol + k);
    v2f b1 = *(const v2f*)(wcol + 16 * FEAT + k);
    v2f b2 = *(const v2f*)(wcol + 32 * FEAT + k);
    v2f b3 = *(const v2f*)(wcol + 48 * FEAT + k);
    acc[0] = __builtin_amdgcn_wmma_f32_16x16x4_f32(false, a, false, b0, (short)0, acc[0], false, false);
    acc[1] = __builtin_amdgcn_wmma_f32_16x16x4_f32(false, a, false, b1, (short)0, acc[1], false, false);
    acc[2] = __builtin_amdgcn_wmma_f32_16x16x4_f32(false, a, false, b2, (short)0, acc[2], false, false);
    acc[3] = __builtin_amdgcn_wmma_f32_16x16x4_f32(false, a, false, b3, (short)0, acc[3], false, false);
  }

  #pragma unroll
  for (int t = 0; t < 4; ++t) {
    int n = col0 + 16 * t + l;
    float bv = bias[n];
    #pragma unroll
    for (int r = 0; r < 8; ++r) {
      int m = row0 + r + 8 * half;
      y1[(size_t)m * HID + n] = fmaxf(acc[t][r] + bv, 0.f);
    }
  }
}

// ---------------------------------------------------------------------------
// Column mean / biased variance over batch (BatchNorm training stats).
// ---------------------------------------------------------------------------
__global__ void col_stats(const float* __restrict__ data, int rows, int cols,
                          float* __restrict__ mean, float* __restrict__ var) {
  __shared__ float s1[256];
  __shared__ float s2[256];
  int c = blockIdx.x, tid = threadIdx.x;
  float a = 0.f, b = 0.f;
  for (int r = tid; r < rows; r += 256) {
    float v = data[(size_t)r * cols + c];
    a += v; b += v * v;
  }
  s1[tid] = a; s2[tid] = b;
  __syncthreads();
  for (int s = 128; s > 0; s >>= 1) {
    if (tid < s) { s1[tid] += s1[tid + s]; s2[tid] += s2[tid + s]; }
    __syncthreads();
  }
  if (tid == 0) {
    float m = s1[0] / rows;
    mean[c] = m;
    var[c]  = s2[0] / rows - m * m;
  }
}

// ---------------------------------------------------------------------------
// Fold BN1 affine into FC2 weights.
// ---------------------------------------------------------------------------
__global__ void fold_bn1(const float* __restrict__ w2, const float* __restrict__ b2,
                         const float* __restrict__ gamma1, const float* __restrict__ beta1,
                         const float* __restrict__ mean1, const float* __restrict__ var1,
                         float* __restrict__ w2p, float* __restrict__ b2p) {
  __shared__ float red[256];
  int c = blockIdx.x, tid = threadIdx.x;
  float acc = 0.f;
  for (int j = tid; j < HID; j += 256) {
    float a  = gamma1[j] * rsqrtf(var1[j] + BN_EPS);
    float sh = beta1[j] - a * mean1[j];
    float wv = w2[c * HID + j];
    w2p[c * HID + j] = wv * a;
    acc += wv * sh;
  }
  red[tid] = acc;
  __syncthreads();
  for (int s = 128; s > 0; s >>= 1) {
    if (tid < s) red[tid] += red[tid + s];
    __syncthreads();
  }
  if (tid == 0) b2p[c] = b2[c] + red[0];
}

// ---------------------------------------------------------------------------
// FC2: r2 = relu(y1 @ w2'^T + b2'). One block per batch row.
// ---------------------------------------------------------------------------
__global__ void fc2_kernel(const float* __restrict__ y1, const float* __restrict__ w2p,
                           const float* __restrict__ b2p, float* __restrict__ r2) {
  __shared__ float sh[256 * NCLS];
  int row = blockIdx.x, tid = threadIdx.x;
  float acc[NCLS];
  #pragma unroll
  for (int c = 0; c < NCLS; ++c) acc[c] = 0.f;
  const float* yr = y1 + (size_t)row * HID;
  for (int j = tid; j < HID; j += 256) {
    float y = yr[j];
    #pragma unroll
    for (int c = 0; c < NCLS; ++c) acc[c] += y * w2p[c * HID + j];
  }
  #pragma unroll
  for (int c = 0; c < NCLS; ++c) sh[tid * NCLS + c] = acc[c];
  __syncthreads();
  for (int s = 128; s > 0; s >>= 1) {
    if (tid < s) {
      #pragma unroll
      for (int c = 0; c < NCLS; ++c) sh[tid * NCLS + c] += sh[(tid + s) * NCLS + c];
    }
    __syncthreads();
  }
  if (tid < NCLS) r2[(size_t)row * NCLS + tid] = fmaxf(sh[tid] + b2p[tid], 0.f);
}

// ---------------------------------------------------------------------------
// Tiny prep: Q = tril(L) tril(L)^T + eps*I ;  h = G z0 + s0
// ---------------------------------------------------------------------------
__global__ void qp_prep(const float* __restrict__ L, const float* __restrict__ G,
                        const float* __restrict__ z0, const float* __restrict__ s0,
                        float* __restrict__ Qm, float* __restrict__ hv) {
  int tid = threadIdx.x;
  if (tid < NCLS * NCLS) {
    int i = tid / NCLS, j = tid % NCLS;
    int kmax = i < j ? i : j;
    float s = 0.f;
    for (int k = 0; k <= kmax; ++k) s += L[i * NCLS + k] * L[j * NCLS + k];
    if (i == j) s += QP_EPS;
    Qm[tid] = s;
  }
  if (tid < NINEQ) {
    float s = s0[tid];
    for (int c = 0; c < NCLS; ++c) s += G[tid * NCLS + c] * z0[c];
    hv[tid] = s;
  }
}

// ---------------------------------------------------------------------------
// Batched primal-dual interior point.
// Lane = batch element (32/block); wave = 1/4 of the 200 constraints.
// Partial sums (mu, (lam+t)@G, packed KKT) combined via ds_add_f32 atomics;
// 10x10 Cholesky recomputed redundantly by all 4 waves (bitwise identical).
// lam/s statically partitioned per wave -> never shared.
// ---------------------------------------------------------------------------
#define TIDX(i, j) ((i) * ((i) + 1) / 2 + (j))

__global__ __launch_bounds__(128) void qp_solve_kernel(
    const float* __restrict__ r2, const float* __restrict__ gamma2,
    const float* __restrict__ beta2, const float* __restrict__ mean2,
    const float* __restrict__ var2, const float* __restrict__ G,
    const float* __restrict__ Qm, const float* __restrict__ hv,
    float* __restrict__ out) {
  __shared__ float lamS[NINEQ * 32];       // 25600 B
  __shared__ float sS[NINEQ * 32];         // 25600 B
  __shared__ float Qs[NCLS * NCLS];        //   400 B
  __shared__ float hs[NINEQ];              //   800 B
  __shared__ float musT[32];               //   128 B
  __shared__ float uT[32][NCLS];           //  1280 B
  __shared__ float kkT[32][55];            //  7040 B
  __shared__ float minB[NPART][32];        //   512 B   (total 61360 B)

  const int tid = threadIdx.x;
  const int e   = tid & 31;     // element lane
  const int q   = tid >> 5;     // constraint part (wave)
  const int m0  = q * MPART;
  const int m1  = m0 + MPART;
  const int b   = blockIdx.x * 32 + e;

  for (int i = tid; i < NCLS * NCLS; i += 128) Qs[i] = Qm[i];
  for (int i = tid; i < NINEQ; i += 128) hs[i] = hv[i];

  float z[NCLS], p[NCLS];
  #pragma unroll
  for (int c = 0; c < NCLS; ++c) {
    z[c] = 0.f;
    float v = r2[(size_t)b * NCLS + c];
    p[c] = gamma2[c] * (v - mean2[c]) * rsqrtf(var2[c] + BN_EPS) + beta2[c];
  }
  for (int m = m0; m < m1; ++m) { lamS[m * 32 + e] = 1.f; sS[m * 32 + e] = 1.f; }
  __syncthreads();

  for (int it = 0; it < QP_ITERS; ++it) {
    // ---- zero per-element total buffers (part 0 only) ----
    if (q == 0) {
      musT[e] = 0.f;
      #pragma unroll
      for (int n = 0; n < NCLS; ++n) uT[e][n] = 0.f;
      #pragma unroll
      for (int i = 0; i < 55; ++i) kkT[e][i] = 0.f;
    }
    __syncthreads();

    // ---- mu = 0.1 * mean(lam*s) ----
    float pm = 0.f;
    for (int m = m0; m < m1; ++m) pm += lamS[m * 32 + e] * sS[m * 32 + e];
    atomicAdd(&musT[e], pm);
    __syncthreads();
    float mu = 0.1f * musT[e] / (float)NINEQ;

    // ---- main constraint sweep (my 50 m): partials for rhs & KKT ----
    float uL[NCLS];
    #pragma unroll
    for (int n = 0; n < NCLS; ++n) uL[n] = 0.f;
    float kkL[55];
    #pragma unroll
    for (int i = 0; i < 55; ++i) kkL[i] = 0.f;

    for (int m = m0; m < m1; ++m) {
      const float* gm = G + m * NCLS;
      float g[NCLS];
      #pragma unroll
      for (int n = 0; n < NCLS; ++n) g[n] = gm[n];
      float lam = lamS[m * 32 + e], sv = sS[m * 32 + e];
      float sinv = 1.0f / sv;
      float gz = 0.f;
      #pragma unroll
      for (int n = 0; n < NCLS; ++n) gz += g[n] * z[n];
      float rp = gz + sv - hs[m];
      float rc = lam * sv - mu;
      float w  = lam * sinv;
      float t  = (lam * rp - rc) * sinv;
      float lt = lam + t;
      #pragma unroll
      for (int n = 0; n < NCLS; ++n) uL[n] += lt * g[n];
      #pragma unroll
      for (int i = 0; i < NCLS; ++i) {
        #pragma unroll
        for (int j = 0; j <= i; ++j) kkL[TIDX(i, j)] += w * g[i] * g[j];
      }
    }
    #pragma unroll
    for (int n = 0; n < NCLS; ++n) atomicAdd(&uT[e][n], uL[n]);
    #pragma unroll
    for (int i = 0; i < 55; ++i) atomicAdd(&kkT[e][i], kkL[i]);
    __syncthreads();

    // ---- build totals; Cholesky + solve (redundant in all 4 waves) ----
    float rhsv[NCLS];
    #pragma unroll
    for (int n = 0; n < NCLS; ++n) {
      float s = p[n];
      #pragma unroll
      for (int i = 0; i < NCLS; ++i) s += Qs[n * NCLS + i] * z[i];
      rhsv[n] = -(s + uT[e][n]);
    }
    float KK[55];
    #pragma unroll
    for (int i = 0; i < NCLS; ++i) {
      #pragma unroll
      for (int j = 0; j <= i; ++j) KK[TIDX(i, j)] = kkT[e][TIDX(i, j)] + Qs[i * NCLS + j];
    }
    #pragma unroll
    for (int j = 0; j < NCLS; ++j) {
      float d = KK[TIDX(j, j)];
      #pragma unroll
      for (int k = 0; k < j; ++k) { float v = KK[TIDX(j, k)]; d -= v * v; }
      d = sqrtf(d);
      KK[TIDX(j, j)] = d;
      float dinv = 1.f / d;
      #pragma unroll
      for (int i = j + 1; i < NCLS; ++i) {
        float v = KK[TIDX(i, j)];
        #pragma unroll
        for (int k = 0; k < j; ++k) v -= KK[TIDX(i, k)] * KK[TIDX(j, k)];
        KK[TIDX(i, j)] = v * dinv;
      }
    }
    #pragma unroll
    for (int i = 0; i < NCLS; ++i) {
      float v = rhsv[i];
      #pragma unroll
      for (int k = 0; k < i; ++k) v -= KK[TIDX(i, k)] * rhsv[k];
      rhsv[i] = v / KK[TIDX(i, i)];
    }
    #pragma unroll
    for (int ii = NCLS - 1; ii >= 0; --ii) {
      float v = rhsv[ii];
      #pragma unroll
      for (int k = ii + 1; k < NCLS; ++k) v -= KK[TIDX(k, ii)] * rhsv[k];
      rhsv[ii] = v / KK[TIDX(ii, ii)];
    }
    // rhsv now holds dz

    // ---- fraction-to-boundary step (partial min over my 50 m) ----
    float minr = 1e9f;
    for (int m = m0; m < m1; ++m) {
      const float* gm = G + m * NCLS;
      float g[NCLS];
      #pragma unroll
      for (int n = 0; n < NCLS; ++n) g[n] = gm[n];
      float lam = lamS[m * 32 + e], sv = sS[m * 32 + e];
      float sinv = 1.0f / sv, linv = 1.0f / lam;
      float gz = 0.f, gdz = 0.f;
      #pragma unroll
      for (int n = 0; n < NCLS; ++n) { gz += g[n] * z[n]; gdz += g[n] * rhsv[n]; }
      float rp = gz + sv - hs[m];
      float rc = lam * sv - mu;
      float dlam = lam * sinv * (gdz + rp) - rc * sinv;
      float ds   = -(rc + sv * dlam) * linv;
      if (dlam < 0.f) minr = fminf(minr, -lam / dlam);
      if (ds   < 0.f) minr = fminf(minr, -sv  / ds);
    }
    minB[q][e] = minr;
    __syncthreads();
    float mr = minB[0][e];
    #pragma unroll
    for (int qq = 1; qq < NPART; ++qq) mr = fminf(mr, minB[qq][e]);
    float alpha = fminf(1.0f, 0.99f * mr);

    // ---- update lam/s (my range) and z (redundant, identical) ----
    for (int m = m0; m < m1; ++m) {
      const float* gm = G + m * NCLS;
      float g[NCLS];
      #pragma unroll
      for (int n = 0; n < NCLS; ++n) g[n] = gm[n];
      float lam = lamS[m * 32 + e], sv = sS[m * 32 + e];
      float sinv = 1.0f / sv, linv = 1.0f / lam;
      float gz = 0.f, gdz = 0.f;
      #pragma unroll
      for (int n = 0; n < NCLS; ++n) { gz += g[n] * z[n]; gdz += g[n] * rhsv[n]; }
      float rp = gz + sv - hs[m];
      float rc = lam * sv - mu;
      float dlam = lam * sinv * (gdz + rp) - rc * sinv;
      float ds   = -(rc + sv * dlam) * linv;
      lamS[m * 32 + e] = lam + alpha * dlam;
      sS[m * 32 + e]   = sv  + alpha * ds;
    }
    #pragma unroll
    for (int n = 0; n < NCLS; ++n) z[n] += alpha * rhsv[n];
  }

  // ---- log_softmax (part 0 writes) ----
  if (q == 0) {
    float mx = z[0];
    #pragma unroll
    for (int c = 1; c < NCLS; ++c) mx = fmaxf(mx, z[c]);
    float se = 0.f;
    #pragma unroll
    for (int c = 0; c < NCLS; ++c) se += expf(z[c] - mx);
    float lse = mx + logf(se);
    #pragma unroll
    for (int c = 0; c < NCLS; ++c) out[(size_t)b * NCLS + c] = z[c] - lse;
  }
}

// ---------------------------------------------------------------------------
extern "C" void kernel_launch(void* const* d_in, const int* in_sizes, int n_in,
                              void* d_out, int out_size, void* d_ws, size_t ws_size,
                              hipStream_t stream) {
  (void)in_sizes; (void)n_in; (void)out_size; (void)ws_size;
  const float* x      = (const float*)d_in[0];
  const float* fc1_w  = (const float*)d_in[1];
  const float* fc1_b  = (const float*)d_in[2];
  const float* gamma1 = (const float*)d_in[3];
  const float* beta1  = (const float*)d_in[4];
  const float* fc2_w  = (const float*)d_in[5];
  const float* fc2_b  = (const float*)d_in[6];
  const float* gamma2 = (const float*)d_in[7];
  const float* beta2  = (const float*)d_in[8];
  const float* L      = (const float*)d_in[9];
  const float* G      = (const float*)d_in[10];
  const float* z0     = (const float*)d_in[11];
  const float* s0     = (const float*)d_in[12];
  float* out = (float*)d_out;

  float* ws    = (float*)d_ws;
  float* y1    = ws; ws += (size_t)BATCH * HID;
  float* r2    = ws; ws += (size_t)BATCH * NCLS;
  float* mean1 = ws; ws += HID;
  float* var1  = ws; ws += HID;
  float* w2p   = ws; ws += NCLS * HID;
  float* b2p   = ws; ws += NCLS;
  float* mean2 = ws; ws += NCLS;
  float* var2  = ws; ws += NCLS;
  float* Qm    = ws; ws += NCLS * NCLS;
  float* hv    = ws; ws += NINEQ;

  dim3 g1(BATCH / 16, HID / 64);
  fc1_wmma<<<g1, 32, 0, stream>>>(x, fc1_w, fc1_b, y1);
  col_stats<<<HID, 256, 0, stream>>>(y1, BATCH, HID, mean1, var1);
  fold_bn1<<<NCLS, 256, 0, stream>>>(fc2_w, fc2_b, gamma1, beta1, mean1, var1, w2p, b2p);
  fc2_kernel<<<BATCH, 256, 0, stream>>>(y1, w2p, b2p, r2);
  col_stats<<<NCLS, 256, 0, stream>>>(r2, BATCH, NCLS, mean2, var2);
  qp_prep<<<1, 256, 0, stream>>>(L, G, z0, s0, Qm, hv);
  qp_solve_kernel<<<BATCH / 32, 128, 0, stream>>>(r2, gamma2, beta2, mean2, var2, G, Qm, hv, out);
}